// ChamferDistance_37056977829910
// MI455X (gfx1250) — compile-verified
//
#include <hip/hip_runtime.h>
#include <hip/hip_bf16.h>

// CDNA5 / gfx1250 chamfer distance.
// Squared pairwise distances via Gram identity on V_WMMA_F32_16X16X4_F32.
// Trick: K=4 but C=3, so the spare K-slot carries the column term:
//   A row m = (-2x, -2y, -2z, 1),  B col n = (x, y, z, y2[n])
//   => D[m][n] = y2[n] - 2 * (x_m . y_n)         (C = inline 0)
// The row term x2[m] is constant per row and max(.,0) commutes with min, so
// both are applied once per row after the full column sweep:
//   dist[m] = max(x2[m] + min_n D[m][n], 0)

typedef __attribute__((ext_vector_type(2))) float v2f;
typedef __attribute__((ext_vector_type(8))) float v8f;

#define WAVES_PER_BLOCK 8
#define ROWS_PER_BLOCK (WAVES_PER_BLOCK * 16)

__global__ __launch_bounds__(256) void chamfer_rowmin_kernel(
    const float* __restrict__ pts1, const float* __restrict__ pts2,
    float* __restrict__ dmin, int N)
{
    const int b   = blockIdx.y;   // batch
    const int dir = blockIdx.z;   // 0: dist1 (p1 vs p2), 1: dist2 (p2 vs p1)
    const float* __restrict__ Apts = (dir == 0 ? pts1 : pts2) + (size_t)b * N * 3;
    const float* __restrict__ Bpts = (dir == 0 ? pts2 : pts1) + (size_t)b * N * 3;

    const int lane = threadIdx.x & 31;
    const int wave = threadIdx.x >> 5;
    const int half = lane >> 4;     // 0: K=0,1 / rows M=r ; 1: K=2,3 / rows M=r+8
    const int m    = lane & 15;

    const int rowBase = blockIdx.x * ROWS_PER_BLOCK + wave * 16;

    // ---- Load this wave's 16 A-points (both lane-halves load the same point m)
    const float* ap = Apts + (size_t)(rowBase + m) * 3;
    const float ax = ap[0], ay = ap[1], az = ap[2];
    const float x2 = ax * ax + ay * ay + az * az;

    // A fragment (16x4 f32, MxK): lane m, half0 -> (K0,K1)=(-2x,-2y);
    // half1 -> (K2,K3)=(-2z, 1).  The 1 in K3 picks up y2 from the B fragment.
    v2f afrag;
    afrag.x = half ? (-2.0f * az) : (-2.0f * ax);
    afrag.y = half ? 1.0f         : (-2.0f * ay);

    // x2 for the 8 rows this lane owns in the C/D layout: row = r + 8*half
    float x2row[8];
#pragma unroll
    for (int r = 0; r < 8; ++r)
        x2row[r] = __shfl(x2, r + 8 * half, 32);

    float rowmin[8];
#pragma unroll
    for (int r = 0; r < 8; ++r) rowmin[r] = 3.402823466e38f;

    const v8f czero = {};

    // ---- Sweep all B points, 16 columns per WMMA tile
#pragma unroll 4
    for (int cb = 0; cb < N; cb += 16) {
        const float* bp = Bpts + (size_t)(cb + m) * 3;
        const float bx = bp[0], by = bp[1], bz = bp[2];
        const float y2 = bx * bx + by * by + bz * bz;   // col term for col (lane&15)

        // B fragment (4x16 f32, KxN): lane n, half0 -> (K0,K1)=(x,y);
        // half1 -> (K2,K3)=(z, y2)
        v2f bfrag;
        bfrag.x = half ? bz : bx;
        bfrag.y = half ? y2 : by;

        // acc[r] = y2[col] - 2 * dot(A_row(r+8*half), B_col(lane&15))
        v8f acc = __builtin_amdgcn_wmma_f32_16x16x4_f32(
            /*neg_a=*/false, afrag, /*neg_b=*/false, bfrag,
            /*c_mod=*/(short)0, czero, /*reuse_a=*/false, /*reuse_b=*/false);

#pragma unroll
        for (int r = 0; r < 8; ++r)
            rowmin[r] = fminf(rowmin[r], acc[r]);
    }

    // ---- Min across the 16 lanes of each half (cols), per owned row, then
    //      add the deferred row term x2 and apply the deferred max(.,0) clamp.
#pragma unroll
    for (int r = 0; r < 8; ++r) {
        float v = rowmin[r];
        v = fminf(v, __shfl_xor(v, 1, 32));
        v = fminf(v, __shfl_xor(v, 2, 32));
        v = fminf(v, __shfl_xor(v, 4, 32));
        v = fminf(v, __shfl_xor(v, 8, 32));
        rowmin[r] = fmaxf(x2row[r] + v, 0.0f);
    }

    // lane 0 holds rows rowBase+0..7, lane 16 holds rows rowBase+8..15
    if (m == 0) {
        float* out = dmin + ((size_t)(dir * 4 + b)) * N + rowBase + 8 * half;
#pragma unroll
        for (int r = 0; r < 8; ++r) out[r] = rowmin[r];
    }
}

// loss = sum_b( mean(dist1_b) + mean(dist2_b) ) = sum(all mins) / N
__global__ __launch_bounds__(256) void chamfer_reduce_kernel(
    const float* __restrict__ dmin, float* __restrict__ out, int total, float scale)
{
    __shared__ float smem[256];
    float acc = 0.0f;
    for (int i = threadIdx.x; i < total; i += 256) acc += dmin[i];
    smem[threadIdx.x] = acc;
    __syncthreads();
    for (int stride = 128; stride > 0; stride >>= 1) {
        if ((int)threadIdx.x < stride) smem[threadIdx.x] += smem[threadIdx.x + stride];
        __syncthreads();
    }
    if (threadIdx.x == 0) out[0] = smem[0] * scale;
}

extern "C" void kernel_launch(void* const* d_in, const int* in_sizes, int n_in,
                              void* d_out, int out_size, void* d_ws, size_t ws_size,
                              hipStream_t stream) {
    const float* p1 = (const float*)d_in[0];   // [4, N, 3] fp32
    const float* p2 = (const float*)d_in[1];   // [4, N, 3] fp32
    const int N = in_sizes[0] / (4 * 3);       // 8192

    float* dmin = (float*)d_ws;                // [2, 4, N] fp32 (fully overwritten)

    dim3 grid(N / ROWS_PER_BLOCK, 4, 2);       // 64 x 4 x 2 blocks, 8 waves each
    chamfer_rowmin_kernel<<<grid, 256, 0, stream>>>(p1, p2, dmin, N);

    const int total = 2 * 4 * N;
    chamfer_reduce_kernel<<<1, 256, 0, stream>>>(dmin, (float*)d_out, total,
                                                 1.0f / (float)N);
}